// GCN_18820546691595
// MI455X (gfx1250) — compile-verified
//
#include <hip/hip_runtime.h>

typedef __attribute__((ext_vector_type(2))) float v2f;
typedef __attribute__((ext_vector_type(8))) float v8f;

#define IN_CH 128
#define HID 32

// ---------------- elementwise / setup kernels ----------------

__global__ void init_deg_kernel(float* __restrict__ deg, int n) {
  int i = blockIdx.x * blockDim.x + threadIdx.x;
  if (i < n) deg[i] = 1.0f;  // self-loop contributes 1 to in-degree
}

__global__ void degree_kernel(const int* __restrict__ dst, float* __restrict__ deg, int e_count) {
  int e = blockIdx.x * blockDim.x + threadIdx.x;
  if (e < e_count) atomicAdd(&deg[dst[e]], 1.0f);
}

__global__ void dinv_kernel(float* __restrict__ degdinv, int n) {
  int i = blockIdx.x * blockDim.x + threadIdx.x;
  if (i < n) {
    float d = degdinv[i];
    degdinv[i] = (d > 0.0f) ? rsqrtf(d) : 0.0f;
  }
}

// agg[i][c] = bias[c] + h[i][c] * dinv[i]^2   (self-loop edge + bias, pre-scatter)
__global__ void init_agg_kernel(const float* __restrict__ h, const float* __restrict__ dinv,
                                const float* __restrict__ bias, float* __restrict__ agg, int n) {
  int i = blockIdx.x * blockDim.x + threadIdx.x;
  if (i < n * HID) {
    int node = i >> 5;
    int c = i & 31;
    float di = dinv[node];
    agg[i] = bias[c] + h[i] * di * di;
  }
}

__global__ void relu_kernel(const float* __restrict__ in, float* __restrict__ out, int total) {
  int i = blockIdx.x * blockDim.x + threadIdx.x;
  if (i < total) out[i] = fmaxf(in[i], 0.0f);
}

// One wave32 per edge, lane == channel (HID == 32 == wave width).
// Gather h[src] is a fully coalesced 128B line; scatter is global_atomic_add_f32.
// h and agg (12.8 MB each) are L2-resident on MI455X (192 MB L2).
__global__ __launch_bounds__(256)
void edge_agg_kernel(const int* __restrict__ src, const int* __restrict__ dst,
                     const float* __restrict__ dinv, const float* __restrict__ h,
                     float* __restrict__ agg, int e_count) {
  int lane = threadIdx.x & 31;
  int e = blockIdx.x * (blockDim.x >> 5) + (threadIdx.x >> 5);
  if (e >= e_count) return;
  int s = src[e];
  int d = dst[e];
  float nrm = dinv[s] * dinv[d];
  float v = h[s * HID + lane] * nrm;
  atomicAdd(&agg[d * HID + lane], v);
}

// ---------------- WMMA fp32 GEMM: out[M x 32] = A[M x K] @ W[K x 32] ----------------
// Block = 128 threads = 4 waves. Block computes a 32x32 output tile:
//   wave (wid>>1) -> M half (0/16), wave (wid&1) -> N half (0/16).
// A tile staged in LDS (padded row stride K+1 to avoid 16-way bank conflicts).
// Uses V_WMMA_F32_16X16X4_F32 (full fp32, matches fp32 reference precision).
template <int K>
__global__ __launch_bounds__(128)
void gemm_n32_kernel(const float* __restrict__ A, const float* __restrict__ W,
                     float* __restrict__ out, int M) {
  constexpr int LDK = K + 1;
  __shared__ float sA[32 * LDK];

  const int tid = threadIdx.x;
  const int block_row0 = blockIdx.x * 32;

  // Cooperative, coalesced load of the 32 x K A-tile into LDS.
  for (int idx = tid; idx < 32 * K; idx += 128) {
    int r = idx / K;
    int c = idx - r * K;
    int gr = block_row0 + r;
    sA[r * LDK + c] = (gr < M) ? A[(long long)gr * K + c] : 0.0f;
  }
  __syncthreads();

  const int wid   = tid >> 5;
  const int lane  = tid & 31;
  const int lhalf = lane >> 4;   // 0: lanes 0-15, 1: lanes 16-31
  const int l15   = lane & 15;
  const int warp_m = (wid >> 1) * 16;  // row offset within 32-row tile
  const int n0     = (wid & 1) * 16;   // col offset within 32 cols

  v8f acc = {};
  for (int k = 0; k < K; k += 4) {
    // A frag (16x4 fp32, ISA layout): lane<16 holds K=k,k+1 ; lane>=16 holds K=k+2,k+3
    const float* ap = &sA[(warp_m + l15) * LDK + k + lhalf * 2];
    v2f a;
    a.x = ap[0];
    a.y = ap[1];
    // B frag (4x16 fp32): row k striped across lanes; lane<16 -> K=k,k+1 ; lane>=16 -> K=k+2,k+3
    const float* bp = &W[(k + lhalf * 2) * HID + n0 + l15];
    v2f b;
    b.x = bp[0];
    b.y = bp[HID];
    // 8 args: (neg_a, A, neg_b, B, c_mod, C, reuse_a, reuse_b)
    acc = __builtin_amdgcn_wmma_f32_16x16x4_f32(false, a, false, b, (short)0, acc,
                                                false, false);
  }

  // D layout (16x16 fp32 C/D): VGPR j -> row j (lanes 0-15) / row j+8 (lanes 16-31)
  const int grow_base = block_row0 + warp_m + lhalf * 8;
  const int col = n0 + l15;
#pragma unroll
  for (int j = 0; j < 8; ++j) {
    int gr = grow_base + j;
    if (gr < M) out[(long long)gr * HID + col] = acc[j];
  }
}

// ---------------- launch ----------------

extern "C" void kernel_launch(void* const* d_in, const int* in_sizes, int n_in,
                              void* d_out, int out_size, void* d_ws, size_t ws_size,
                              hipStream_t stream) {
  const float* x  = (const float*)d_in[0];
  const int*   ei = (const int*)d_in[1];   // [2, E] row-major: row0 = src, row1 = dst
  const float* W1 = (const float*)d_in[2];
  const float* b1 = (const float*)d_in[3];
  const float* W2 = (const float*)d_in[4];
  const float* b2 = (const float*)d_in[5];
  float* out = (float*)d_out;

  const int N = in_sizes[0] / IN_CH;   // 100000
  const int E = in_sizes[1] / 2;       // 3200000
  const int* src = ei;
  const int* dst = ei + E;

  // workspace layout: [dinv (N, padded)] [bufA (N*32)] [bufB (N*32)]
  float* ws = (float*)d_ws;
  const int Npad = (N + 63) & ~63;
  float* dinv = ws;                 // deg, then dinv in place
  float* bufA = ws + Npad;          // h1, later relu(h), i.e. layer-2 input
  float* bufB = bufA + (size_t)N * HID;  // agg1, later h2

  const int T = 256;
  const unsigned gN    = (N + T - 1) / T;
  const unsigned gE    = (E + T - 1) / T;
  const unsigned gNC   = ((unsigned)N * HID + T - 1) / T;
  const unsigned gEdge = (E + (T / 32) - 1) / (T / 32);   // 8 edges / block
  const unsigned gGemm = (N + 31) / 32;

  // degree (with self-loops) -> dinv
  init_deg_kernel<<<gN, T, 0, stream>>>(dinv, N);
  degree_kernel<<<gE, T, 0, stream>>>(dst, dinv, E);
  dinv_kernel<<<gN, T, 0, stream>>>(dinv, N);

  // ---- layer 1 ----
  gemm_n32_kernel<IN_CH><<<gGemm, 128, 0, stream>>>(x, W1, bufA, N);        // h1 = x @ W1
  init_agg_kernel<<<gNC, T, 0, stream>>>(bufA, dinv, b1, bufB, N);          // bias + self-loop
  edge_agg_kernel<<<gEdge, T, 0, stream>>>(src, dst, dinv, bufA, bufB, E);  // scatter-add
  relu_kernel<<<gNC, T, 0, stream>>>(bufB, bufA, N * HID);                  // bufA = relu(agg1)

  // ---- layer 2 ----
  gemm_n32_kernel<HID><<<gGemm, 128, 0, stream>>>(bufA, W2, bufB, N);       // h2 = relu @ W2
  init_agg_kernel<<<gNC, T, 0, stream>>>(bufB, dinv, b2, out, N);           // bias + self-loop
  edge_agg_kernel<<<gEdge, T, 0, stream>>>(src, dst, dinv, bufB, out, E);   // scatter-add
}